// SCARF_14250701488480
// MI455X (gfx1250) — compile-verified
//
#include <hip/hip_runtime.h>
#include <math.h>

// SCARF transform-blend: brute-force KNN via V_WMMA_F32_16X16X4_F32 with
// TDM (tensor_load_to_lds) staging, then confidence-gated transform blend.
//
// WMMA mapping (D = A*B + C, 16x16x4 f32):
//   A row m (template vert): (-2x, -2y, -2z, |t|^2)   <- precomputed by prep
//   B col n (query point)  : ( px,  py,  pz, 1   )
//   C[m][n] = |t_m|^2 - 2<t_m, p_n>   (|p_n|^2 is constant per column: added
//                                      after selection, ordering unaffected)

typedef __attribute__((ext_vector_type(2))) float v2f;
typedef __attribute__((ext_vector_type(8))) float v8f;
typedef __attribute__((ext_vector_type(4))) unsigned int u32x4;
typedef __attribute__((ext_vector_type(8))) int i32x8;
typedef __attribute__((ext_vector_type(4))) int i32x4;

#define JNT   55
#define KNN   6
#define CHUNK 2048          // verts per LDS chunk (32 KB of float4)
#define BLK   128           // 4 wave32
#define PTS_PER_BLK 64      // 16 points per wave
#define BIGF  3.0e38f

#if defined(__has_builtin)
#if __has_builtin(__builtin_amdgcn_tensor_load_to_lds)
#define HAVE_TDM 1
#endif
#endif

// ---------------------------------------------------------------- prep ----
__global__ __launch_bounds__(256) void scarf_prep_kernel(
    const float* __restrict__ tmpl,    // [V,3]
    float4* __restrict__ wsA,          // [Vpad] fragment-ready rows
    int V, int Vpad)
{
    const int t = blockIdx.x * 256 + threadIdx.x;
    if (t >= Vpad) return;
    float4 o;
    if (t < V) {
        const float x = tmpl[t*3+0], y = tmpl[t*3+1], z = tmpl[t*3+2];
        o = make_float4(-2.0f*x, -2.0f*y, -2.0f*z, x*x + y*y + z*z);
    } else {
        o = make_float4(0.0f, 0.0f, 0.0f, BIGF);   // pad row: never selected
    }
    wsA[t] = o;
}

// ----------------------------------------------------------------- knn ----
__global__ __launch_bounds__(BLK) void scarf_knn_kernel(
    const float* __restrict__ points,   // [N,3]
    const float* __restrict__ wsA,      // [Vpad*4] fragment-ready
    float* __restrict__ outD,           // [N*KNN]
    int*   __restrict__ outI,           // [N*KNN]
    int N, int Vpad)
{
    __shared__ float sA[CHUNK * 4] __attribute__((aligned(16)));
    __shared__ float sMV[4][32][KNN];
    __shared__ int   sMI[4][32][KNN];

    const int  tid  = threadIdx.x;
    const int  lane = tid & 31;
    const int  wave = tid >> 5;
    const int  col  = lane & 15;
    const bool hi   = lane >= 16;
    const int  sel  = hi ? 2 : 0;       // float offset within a row

    const int p = blockIdx.x * PTS_PER_BLK + wave * 16 + col;

    float px = 0.f, py = 0.f, pz = 0.f;
    if (p < N) { px = points[p*3+0]; py = points[p*3+1]; pz = points[p*3+2]; }
    const float p2 = px*px + py*py + pz*pz;

    v2f bfrag;                          // B fragment (4x16 K x N)
    bfrag.x = hi ? pz : px;
    bfrag.y = hi ? 1.0f : py;

    float bv[KNN]; int bi[KNN];
#pragma unroll
    for (int k = 0; k < KNN; ++k) { bv[k] = BIGF; bi[k] = 0; }

    for (int chunkBase = 0; chunkBase < Vpad; chunkBase += CHUNK) {
        const int cnt = (Vpad - chunkBase) < CHUNK ? (Vpad - chunkBase) : CHUNK;
        // cnt is always a multiple of 16 (Vpad is, CHUNK is).

#ifdef HAVE_TDM
        // ---- TDM: one descriptor copies cnt*16 bytes global->LDS --------
        if (wave == 0) {
            const unsigned long long ga =
                (unsigned long long)(size_t)wsA + (unsigned long long)chunkBase * 16ull;
            const unsigned int ldsOff = (unsigned int)(size_t)(&sA[0]); // addr[31:0] = LDS offset
            const unsigned int nTensor = (unsigned int)Vpad * 4u;       // elems, data_size=4B
            const unsigned int nTile   = (unsigned int)cnt * 4u;        // <= 8192, fits 16b

            u32x4 g0;
            g0[0] = 1u;                                        // count=1 valid D#
            g0[1] = ldsOff;                                    // lds_addr
            g0[2] = (unsigned int)(ga & 0xFFFFFFFFull);        // global_addr[31:0]
            g0[3] = (unsigned int)((ga >> 32) & 0x1FFFFFFull)  // global_addr[56:32]
                    | (2u << 30);                              // type=2 (image)

            i32x8 g1 = {0,0,0,0,0,0,0,0};
            g1[0] = (int)(2u << 16);                           // data_size=4 bytes
            g1[1] = (int)((nTensor & 0xFFFFu) << 16);          // tensor_dim0[15:0]
            g1[2] = (int)((nTensor >> 16) & 0xFFFFu)           // tensor_dim0[31:16]
                    | (int)(1u << 16);                         // tensor_dim1 = 1
            g1[3] = (int)(nTile << 16);                        // tile_dim0
            g1[5] = (int)nTensor;                              // tensor_dim0_stride[31:0]

            i32x4 gz = {0,0,0,0};
#if __clang_major__ >= 23
            i32x8 gz8 = {0,0,0,0,0,0,0,0};
            __builtin_amdgcn_tensor_load_to_lds(g0, g1, gz, gz, gz8, 0);
#else
            __builtin_amdgcn_tensor_load_to_lds(g0, g1, gz, gz, 0);
#endif
            __builtin_amdgcn_s_wait_tensorcnt(0);
        }
        __syncthreads();
#else
        // ---- fallback: cooperative copy -------------------------------
        for (int t = tid; t < cnt; t += BLK)
            ((float4*)sA)[t] = ((const float4*)wsA)[chunkBase + t];
        __syncthreads();
#endif

#pragma unroll 2
        for (int t0 = 0; t0 < cnt; t0 += 16) {
            const int m = t0 + col;
            // A fragment: one aligned 8-byte LDS load per lane.
            const v2f afrag = *(const v2f*)(&sA[m*4 + sel]);

            v8f c = {};
            c = __builtin_amdgcn_wmma_f32_16x16x4_f32(
                    false, afrag, false, bfrag, (short)0, c, false, false);

            const int rowBase = chunkBase + t0 + (hi ? 8 : 0);
#pragma unroll
            for (int r = 0; r < 8; ++r) {
                const float val = c[r];
                const int   vid = rowBase + r;
                if (val < bv[KNN-1]) {
                    int pos = KNN-1;
#pragma unroll
                    for (int q = KNN-2; q >= 0; --q) {
                        if (val < bv[q]) { bv[q+1] = bv[q]; bi[q+1] = bi[q]; pos = q; }
                    }
                    bv[pos] = val; bi[pos] = vid;
                }
            }
        }
        __syncthreads();
    }

    // Merge the two half-lanes of each point.
#pragma unroll
    for (int k = 0; k < KNN; ++k) {
        sMV[wave][lane][k] = bv[k];
        sMI[wave][lane][k] = bi[k];
    }
    __syncthreads();

    if (!hi && p < N) {
        float cv[2*KNN]; int ci[2*KNN];
#pragma unroll
        for (int k = 0; k < KNN; ++k) {
            cv[k]     = sMV[wave][lane][k];      ci[k]     = sMI[wave][lane][k];
            cv[KNN+k] = sMV[wave][lane+16][k];   ci[KNN+k] = sMI[wave][lane+16][k];
        }
#pragma unroll
        for (int k = 0; k < KNN; ++k) {
            int best = 0;
#pragma unroll
            for (int q = 1; q < 2*KNN; ++q)
                if (cv[q] < cv[best]) best = q;
            outD[p*KNN + k] = fmaxf(cv[best] + p2, 0.0f);
            outI[p*KNN + k] = ci[best];
            cv[best] = BIGF;
        }
    }
}

// --------------------------------------------------------------- apply ----
__global__ __launch_bounds__(256) void scarf_apply_kernel(
    const float* __restrict__ lbs,   // [V,55]
    const float* __restrict__ vt,    // [V,16]
    const float* __restrict__ wsD,
    const int*   __restrict__ wsI,
    float* __restrict__ outDist,     // [N]
    float* __restrict__ outXf,       // [N*16]
    int N)
{
    const int p = blockIdx.x * blockDim.x + threadIdx.x;
    if (p >= N) return;

    float d2[KNN]; int idx[KNN];
#pragma unroll
    for (int k = 0; k < KNN; ++k) { d2[k] = wsD[p*KNN+k]; idx[k] = wsI[p*KNN+k]; }

    const float* __restrict__ w0row = lbs + (size_t)idx[0] * JNT;
    float w[KNN];
    w[0] = expf(-d2[0]);
#pragma unroll
    for (int k = 1; k < KNN; ++k) {
        const float* __restrict__ wkrow = lbs + (size_t)idx[k] * JNT;
        float s = 0.0f;
        for (int j = 0; j < JNT; ++j)
            s += fabsf(wkrow[j] - w0row[j]);
        const float conf = (expf(-s * 50.0f) > 0.9f) ? 1.0f : 0.0f;
        w[k] = expf(-d2[k]) * conf;
    }

    float wsum = 0.0f, dsum = 0.0f;
#pragma unroll
    for (int k = 0; k < KNN; ++k) { wsum += w[k]; dsum += w[k] * d2[k]; }
    const float inv = 1.0f / wsum;

    outDist[p] = dsum * inv;

    float4 acc[4] = {make_float4(0,0,0,0), make_float4(0,0,0,0),
                     make_float4(0,0,0,0), make_float4(0,0,0,0)};
#pragma unroll
    for (int k = 0; k < KNN; ++k) {
        const float wk = w[k] * inv;
        const float4* __restrict__ row = (const float4*)(vt + (size_t)idx[k] * 16);
#pragma unroll
        for (int i = 0; i < 4; ++i) {
            float4 r = row[i];
            acc[i].x += wk * r.x; acc[i].y += wk * r.y;
            acc[i].z += wk * r.z; acc[i].w += wk * r.w;
        }
    }
    float4* __restrict__ dst = (float4*)(outXf + (size_t)p * 16);
#pragma unroll
    for (int i = 0; i < 4; ++i) dst[i] = acc[i];
}

// -------------------------------------------------------------- launch ----
extern "C" void kernel_launch(void* const* d_in, const int* in_sizes, int n_in,
                              void* d_out, int out_size, void* d_ws, size_t ws_size,
                              hipStream_t stream) {
    const float* lbs = (const float*)d_in[0];   // [V,55]
    const float* vtf = (const float*)d_in[1];   // [1,V,4,4]
    const float* pts = (const float*)d_in[2];   // [1,N,3]
    const float* tpl = (const float*)d_in[3];   // [1,V,3]
    // d_in[4] is K (==6), baked in as KNN.

    const int V    = in_sizes[0] / JNT;
    const int N    = in_sizes[2] / 3;
    const int Vpad = (V + 15) & ~15;

    // workspace layout: [wsA: Vpad float4][wsD: N*KNN f32][wsI: N*KNN i32]
    float4* wsA = (float4*)d_ws;
    float*  wsD = (float*)(wsA + Vpad);
    int*    wsI = (int*)(wsD + (size_t)N * KNN);

    float* outDist = (float*)d_out;
    float* outXf   = outDist + N;

    scarf_prep_kernel<<<dim3((Vpad + 255) / 256), dim3(256), 0, stream>>>(
        tpl, wsA, V, Vpad);
    scarf_knn_kernel<<<dim3((N + PTS_PER_BLK - 1) / PTS_PER_BLK), dim3(BLK), 0, stream>>>(
        pts, (const float*)wsA, wsD, wsI, N, Vpad);
    scarf_apply_kernel<<<dim3((N + 255) / 256), dim3(256), 0, stream>>>(
        lbs, vtf, wsD, wsI, outDist, outXf, N);
}